// TiedLinear_68917045232330
// MI455X (gfx1250) — compile-verified
//
#include <hip/hip_runtime.h>

typedef float v4f __attribute__((ext_vector_type(4)));
typedef int   v4i __attribute__((ext_vector_type(4)));

// address-space-qualified vector pointers for the async-to-LDS builtin
typedef __attribute__((address_space(1))) v4i g_v4i;   // global
typedef __attribute__((address_space(3))) v4i l_v4i;   // LDS

#define LDS_W 4096   // weight slots staged in LDS (16 KB); F == 4096 in this problem

#if defined(__gfx1250__) && __has_builtin(__builtin_amdgcn_global_load_async_to_lds_b128) && __has_builtin(__builtin_amdgcn_s_wait_asynccnt)
#define USE_ASYNC_LDS 1
#else
#define USE_ASYNC_LDS 0
#endif

// out[b,d] = bias[d] + mask[b,d]  (also overwrites the 0xAA poison before scatter)
__global__ __launch_bounds__(256)
void tied_init_kernel(const float* __restrict__ bias,
                      const float* __restrict__ mask,
                      float* __restrict__ out, int BD, int D) {
  int i = blockIdx.x * blockDim.x + threadIdx.x;
  int e = i * 4;
  if (e + 3 < BD) {
    // D is a multiple of 4, so (e % D) is 16B-aligned within a row
    v4f m = *(const v4f*)(mask + e);
    v4f b = *(const v4f*)(bias + (e % D));
    *(v4f*)(out + e) = m + b;
  } else {
    for (int k = 0; k < 4; ++k) {
      int ee = e + k;
      if (ee < BD) out[ee] = mask[ee] + bias[ee % D];
    }
  }
}

// scatter-reduce: out[b*D+d] += values * weight[f], FP32 atomics resolved in L2
__global__ __launch_bounds__(256)
void tied_scatter_kernel(const float* __restrict__ values,
                         const int*   __restrict__ b_idx,
                         const int*   __restrict__ d_idx,
                         const int*   __restrict__ f_idx,
                         const float* __restrict__ weight,
                         float*       __restrict__ out,
                         int nnz, int D, int F) {
  __shared__ float w_lds[LDS_W];
  const bool use_lds = (F <= LDS_W);   // uniform across the grid

  // ---- stage weight into LDS (CDNA5 async global->LDS path when available) ----
  if (use_lds) {
#if USE_ASYNC_LDS
    for (int j = threadIdx.x * 4; j < F; j += blockDim.x * 4) {
      __builtin_amdgcn_global_load_async_to_lds_b128(
          (g_v4i*)(weight + j),          // global source (AS1, vector-typed)
          (l_v4i*)(&w_lds[j]),           // LDS destination (AS3)
          /*offset=*/0, /*cpol=*/0);
    }
    __builtin_amdgcn_s_wait_asynccnt(0);   // wave's own async loads done
#else
    for (int j = threadIdx.x; j < F; j += blockDim.x) w_lds[j] = weight[j];
#endif
    __syncthreads();                       // LDS visible to all waves in the block
  }

  const int tid    = blockIdx.x * blockDim.x + threadIdx.x;
  const int stride = gridDim.x * blockDim.x;
  const int nvec   = nnz >> 2;

  const v4f* __restrict__ v4  = (const v4f*)values;
  const v4i* __restrict__ b4  = (const v4i*)b_idx;
  const v4i* __restrict__ dd4 = (const v4i*)d_idx;
  const v4i* __restrict__ f4  = (const v4i*)f_idx;

  // Two specializations so the weight gather has a single, known address space:
  //   use_lds path -> ds_load_b32 from w_lds (AS3 only)
  //   fallback     -> global_load from weight (AS1 only)
  // (A ternary select between the two pointers would force generic/FLAT loads.)
  if (use_lds) {
    for (int i = tid; i < nvec; i += stride) {
      int pn = i + stride;
      if (pn < nvec) {                      // prefetch one grid-stride ahead
        __builtin_prefetch(v4  + pn, 0, 1);
        __builtin_prefetch(b4  + pn, 0, 1);
        __builtin_prefetch(dd4 + pn, 0, 1);
        __builtin_prefetch(f4  + pn, 0, 1);
      }
      // one-shot streams: non-temporal, don't evict L2-resident out/weight
      v4f v  = __builtin_nontemporal_load(v4  + i);
      v4i bb = __builtin_nontemporal_load(b4  + i);
      v4i dd = __builtin_nontemporal_load(dd4 + i);
      v4i ff = __builtin_nontemporal_load(f4  + i);
#pragma unroll
      for (int k = 0; k < 4; ++k) {
        int   row = bb[k] * D + dd[k];
        float c   = v[k] * w_lds[ff[k]];    // LDS gather (ds_load_b32)
        __hip_atomic_fetch_add(out + row, c, __ATOMIC_RELAXED, __HIP_MEMORY_SCOPE_AGENT);
      }
    }
    for (int i = (nvec << 2) + tid; i < nnz; i += stride) {
      int   row = b_idx[i] * D + d_idx[i];
      float c   = values[i] * w_lds[f_idx[i]];
      __hip_atomic_fetch_add(out + row, c, __ATOMIC_RELAXED, __HIP_MEMORY_SCOPE_AGENT);
    }
  } else {
    for (int i = tid; i < nvec; i += stride) {
      int pn = i + stride;
      if (pn < nvec) {
        __builtin_prefetch(v4  + pn, 0, 1);
        __builtin_prefetch(b4  + pn, 0, 1);
        __builtin_prefetch(dd4 + pn, 0, 1);
        __builtin_prefetch(f4  + pn, 0, 1);
      }
      v4f v  = __builtin_nontemporal_load(v4  + i);
      v4i bb = __builtin_nontemporal_load(b4  + i);
      v4i dd = __builtin_nontemporal_load(dd4 + i);
      v4i ff = __builtin_nontemporal_load(f4  + i);
#pragma unroll
      for (int k = 0; k < 4; ++k) {
        int   row = bb[k] * D + dd[k];
        float c   = v[k] * weight[ff[k]];   // global gather (L0/L2-resident)
        __hip_atomic_fetch_add(out + row, c, __ATOMIC_RELAXED, __HIP_MEMORY_SCOPE_AGENT);
      }
    }
    for (int i = (nvec << 2) + tid; i < nnz; i += stride) {
      int   row = b_idx[i] * D + d_idx[i];
      float c   = values[i] * weight[f_idx[i]];
      __hip_atomic_fetch_add(out + row, c, __ATOMIC_RELAXED, __HIP_MEMORY_SCOPE_AGENT);
    }
  }
}

extern "C" void kernel_launch(void* const* d_in, const int* in_sizes, int n_in,
                              void* d_out, int out_size, void* d_ws, size_t ws_size,
                              hipStream_t stream) {
  const float* values = (const float*)d_in[0];
  const int*   b_idx  = (const int*)  d_in[1];
  const int*   d_idx  = (const int*)  d_in[2];
  const int*   f_idx  = (const int*)  d_in[3];
  const float* weight = (const float*)d_in[4];
  const float* bias   = (const float*)d_in[5];
  const float* mask   = (const float*)d_in[6];
  float*       out    = (float*)d_out;

  const int nnz = in_sizes[0];
  const int F   = in_sizes[4];
  const int D   = in_sizes[5];
  const int BD  = out_size;

  // Phase 1: out = bias + mask
  {
    const int thr = 256;
    int vec = (BD + 3) / 4;
    int blocks = (vec + thr - 1) / thr;
    tied_init_kernel<<<blocks, thr, 0, stream>>>(bias, mask, out, BD, D);
  }

  // Phase 2: scatter-reduce with L2 atomics, persistent grid-stride
  {
    const int thr = 256;
    long nvec = nnz / 4;
    long need = (nvec + thr - 1) / thr;
    if (need < 1) need = 1;
    int blocks = (int)(need < 4096 ? need : 4096);
    tied_scatter_kernel<<<blocks, thr, 0, stream>>>(values, b_idx, d_idx, f_idx,
                                                    weight, out, nnz, D, F);
  }
}